// mymodel_31026843746696
// MI455X (gfx1250) — compile-verified
//
#include <hip/hip_runtime.h>
#include <math.h>

// ---------------- CDNA5 (gfx1250) wave32 WMMA types ----------------
typedef __attribute__((ext_vector_type(2))) float v2f;
typedef __attribute__((ext_vector_type(8))) float v8f;

#define BN_INV 0.9999950000375f
#define DN   256   // feature dim
#define CCH  12    // cnn channels
#define L1   255   // conv1 length
#define P1L  127
#define L2   126
#define P2L  63
#define L3   62
#define P3L  31
#define FLATN 2652

// LDS layout for the main per-edge kernel (floats)
#define P1S 130          // p1 stride (127 valid + pad for WMMA B reads)
#define P2S 66           // p2 stride (63 valid + pad)
#define P3S 32
#define OFF_BASE1 0                       // 12*256
#define OFF_WEFF  3072                    // 2652
#define OFF_W2F   (OFF_WEFF + FLATN)      // 384 = [16][24], rows 12..15 zero
#define OFF_W3F   (OFF_W2F + 384)         // 384
#define OFF_B2F   (OFF_W3F + 384)         // 16 (12..15 zero)
#define OFF_B3F   (OFF_B2F + 16)          // 16
#define OFF_WA    (OFF_B3F + 16)          // 24
#define OFF_WB    (OFF_WA + 24)           // 24
#define OFF_WC    (OFF_WB + 24)           // 1
#define SHC       (OFF_WC + 1)            // = 6573 shared floats
#define PW        (512 + CCH*P1S + CCH*P2S + CCH*P3S)   // per-wave scratch = 3248
#define WAVES_PER_BLK 2
#define EPW 5                              // edges per wave (amortize const loads)
#define LDS_FLOATS (SHC + WAVES_PER_BLK*PW)

// ---------------- small helper kernels ----------------
__global__ void zero_kernel(float* p, long n) {
    long i = (long)blockIdx.x * blockDim.x + threadIdx.x;
    long stride = (long)gridDim.x * blockDim.x;
    for (; i < n; i += stride) p[i] = 0.f;
}

// scatter-add: agg[dst] += x[src]; deg[dst] += 1   (one block = one edge, 256 feats)
__global__ void scatter_kernel(const int* ei, const float* x, float* agg, float* deg, int Eg) {
    int e = blockIdx.x;
    int f = threadIdx.x;
    int s = ei[e];
    int d = ei[Eg + e];
    atomicAdd(&agg[(size_t)d * DN + f], x[(size_t)s * DN + f]);
    if (f == 0) atomicAdd(&deg[d], 1.0f);
}

__global__ void invdeg_kernel(const float* deg, float* invdeg, int n) {
    int i = blockIdx.x * blockDim.x + threadIdx.x;
    if (i < n) invdeg[i] = 1.0f / fmaxf(deg[i], 1.0f);
}

// pack A = [mean | x] (invdeg folded) and B = [Wl | Wr] so the WMMA loop is branch-free
__global__ void pack_af(const float* __restrict__ agg, const float* __restrict__ invdeg,
                        const float* __restrict__ x, float* __restrict__ af, int nNodes) {
    long i = (long)blockIdx.x * blockDim.x + threadIdx.x;
    if (i >= (long)nNodes * 512) return;
    int m = (int)(i >> 9), k = (int)(i & 511);
    af[i] = (k < DN) ? agg[(size_t)m * DN + k] * invdeg[m] : x[(size_t)m * DN + (k - DN)];
}
__global__ void pack_wf(const float* __restrict__ Wl, const float* __restrict__ Wr,
                        float* __restrict__ wf) {
    int i = blockIdx.x * blockDim.x + threadIdx.x;       // 256*512 threads
    int n = i >> 9, k = i & 511;
    wf[i] = (k < DN) ? Wl[n * DN + k] : Wr[n * DN + (k - DN)];
}

// ---------------- h = relu(af @ wf^T + bl) via V_WMMA_F32_16X16X4_F32 ----------------
// One wave per 16x16 tile; K = 512 in 128 steps of 4; inner loop = 2x b64 load + 1 wmma.
__global__ __launch_bounds__(32) void hgemm_wmma(const float* __restrict__ af,
                                                 const float* __restrict__ wf,
                                                 const float* __restrict__ bl,
                                                 float* __restrict__ h, int nNodes) {
    const int m0 = blockIdx.x * 16, n0 = blockIdx.y * 16;
    const int lane = threadIdx.x;
    const int mi = lane & 15;
    const int sel = lane >> 4;               // 0: K-pair {0,1}; 1: K-pair {2,3}
    int m = m0 + mi; if (m >= nNodes) m = nNodes - 1;   // branchless row clamp
    const float* aRow = af + (size_t)m * 512 + sel * 2;
    const float* bRow = wf + (size_t)(n0 + mi) * 512 + sel * 2;
    v8f acc = {};
#pragma unroll 8
    for (int kt = 0; kt < 128; ++kt) {
        v2f a = *(const v2f*)(aRow + kt * 4);
        v2f b = *(const v2f*)(bRow + kt * 4);
        acc = __builtin_amdgcn_wmma_f32_16x16x4_f32(false, a, false, b, (short)0, acc, false, false);
    }
    // C/D layout: VGPR r -> M = m0 + r + 8*sel, N = n0 + mi
    float bias = bl[n0 + mi];
    if (m0 + 16 <= nNodes) {                 // wave-uniform fast path: clean store clause
        float* hOut = h + (size_t)(m0 + sel * 8) * DN + n0 + mi;
#pragma unroll
        for (int r = 0; r < 8; ++r)
            hOut[(size_t)r * DN] = fmaxf(acc[r] + bias, 0.f);
    } else {                                 // partial last tile
#pragma unroll
        for (int r = 0; r < 8; ++r) {
            int mm = m0 + r + sel * 8;
            if (mm < nNodes) h[(size_t)mm * DN + n0 + mi] = fmaxf(acc[r] + bias, 0.f);
        }
    }
}

// ---------------- top-K pooling (single block) ----------------
__global__ void topk_kernel(const float* __restrict__ h, const float* __restrict__ topkw,
                            float* __restrict__ topkarr, int nNodes) {
    __shared__ float ssc[2048];
    __shared__ float rv[256];
    __shared__ int   ri[256];
    __shared__ float stv[10];
    __shared__ int   sti[10];
    __shared__ float swn;
    int t = threadIdx.x;
    float w = topkw[t];
    rv[t] = w * w; __syncthreads();
    for (int s = 128; s > 0; s >>= 1) { if (t < s) rv[t] += rv[t + s]; __syncthreads(); }
    if (t == 0) swn = sqrtf(rv[0]);
    __syncthreads();
    float inv = 1.0f / swn;
    for (int nI = t; nI < nNodes; nI += 256) {
        const float* row = h + (size_t)nI * DN;
        float acc = 0.f;
        for (int d = 0; d < DN; ++d) acc += row[d] * topkw[d];
        ssc[nI] = acc * inv;
    }
    __syncthreads();
    for (int k = 0; k < 10; ++k) {
        float best = -INFINITY; int bi = 0x7fffffff;
        for (int nI = t; nI < nNodes; nI += 256) {
            float s = ssc[nI];
            if (s > best || (s == best && nI < bi)) { best = s; bi = nI; }
        }
        rv[t] = best; ri[t] = bi; __syncthreads();
        for (int s = 128; s > 0; s >>= 1) {
            if (t < s) {
                if (rv[t + s] > rv[t] || (rv[t + s] == rv[t] && ri[t + s] < ri[t])) {
                    rv[t] = rv[t + s]; ri[t] = ri[t + s];
                }
            }
            __syncthreads();
        }
        if (t == 0) { stv[k] = rv[0]; sti[k] = ri[0]; ssc[ri[0]] = -INFINITY; }
        __syncthreads();
    }
    for (int k = 0; k < 10; ++k)
        topkarr[k * DN + t] = h[(size_t)sti[k] * DN + t] * tanhf(stv[k]);
}

// ---------------- fold BN + shared-topk channels + fc1∘bn4∘fc2 into constants ----------------
__global__ void prep_kernel(const float* __restrict__ topkarr,
        const float* c1w, const float* c1b, const float* c2w, const float* c2b,
        const float* c3w, const float* c3b,
        const float* g1, const float* bb1, const float* g2, const float* bb2,
        const float* g3, const float* bb3, const float* g4, const float* bb4,
        const float* fc1w, const float* fc1b, const float* fc2w, const float* fc2b,
        float* base1, float* wa, float* wb, float* w2f, float* b2f,
        float* w3f, float* b3f, float* weff, float* wconst) {
    __shared__ float s1[12], t1[12], s2[12], t2[12], s3[12], t3[12], s4[128], t4[128];
    int t = threadIdx.x;
    if (t < 12) {
        s1[t] = BN_INV * g1[t]; t1[t] = bb1[t];
        s2[t] = BN_INV * g2[t]; t2[t] = bb2[t];
        s3[t] = BN_INV * g3[t]; t3[t] = bb3[t];
    }
    if (t < 128) { s4[t] = BN_INV * g4[t]; t4[t] = bb4[t]; }
    __syncthreads();
    if (t < 24) {   // per-edge conv1 weights for channels 0 (h[src]) and 1 (h[dst])
        int o = t >> 1, k = t & 1;
        wa[t] = c1w[(o * 12 + 0) * 2 + k] * s1[0];
        wb[t] = c1w[(o * 12 + 1) * 2 + k] * s1[1];
    }
    if (t < 288) {  // conv2/conv3 weights with BN folded, laid out [o][q=i*2+k]
        int o = t / 24, q = t % 24, i = q >> 1, k = q & 1;
        w2f[t] = c2w[(o * 12 + i) * 2 + k] * s2[i];
        w3f[t] = c3w[(o * 12 + i) * 2 + k] * s3[i];
    }
    if (t < 12) {
        float a2 = c2b[t], a3 = c3b[t];
        for (int i = 0; i < 12; ++i)
            for (int k = 0; k < 2; ++k) {
                a2 += c2w[(t * 12 + i) * 2 + k] * t2[i];
                a3 += c3w[(t * 12 + i) * 2 + k] * t3[i];
            }
        b2f[t] = a2; b3f[t] = a3;
    }
    // base1[o][l]: conv1 bias + BN-shift terms + shared topkarr-channel contribution
    for (int idx = t; idx < 12 * L1; idx += 256) {
        int o = idx / L1, l = idx % L1;
        float v = c1b[o];
        for (int i = 0; i < 12; ++i)
            for (int k = 0; k < 2; ++k) {
                float w = c1w[(o * 12 + i) * 2 + k];
                v += w * t1[i];
                if (i >= 2) v += w * s1[i] * topkarr[(i - 2) * DN + l + k];
            }
        base1[idx] = v;
    }
    // weff = (fc2_w ⊙ s4) @ fc1_w ; wconst folds fc1_b, bn4 shift, fc2_b
    for (int f = t; f < FLATN; f += 256) {
        float a = 0.f;
        for (int j = 0; j < 128; ++j) a += fc2w[j] * s4[j] * fc1w[j * FLATN + f];
        weff[f] = a;
    }
    if (t == 0) {
        float a = fc2b[0];
        for (int j = 0; j < 128; ++j) a += fc2w[j] * (s4[j] * fc1b[j] + t4[j]);
        wconst[0] = a;
    }
}

// ---------------- main fused per-edge kernel: conv1 (rank-4) + WMMA conv2/conv3 + folded fc ----------------
__global__ __launch_bounds__(32 * WAVES_PER_BLK)
void edge_cnn_kernel(const int* __restrict__ te, const float* __restrict__ h,
                     const float* __restrict__ base1g, const float* __restrict__ wag,
                     const float* __restrict__ wbg, const float* __restrict__ w2fg,
                     const float* __restrict__ b2fg, const float* __restrict__ w3fg,
                     const float* __restrict__ b3fg, const float* __restrict__ weffg,
                     const float* __restrict__ wconstg, float* __restrict__ out, int E) {
    extern __shared__ float sh[];
    const int tid = threadIdx.x;
    const int lane = tid & 31;
    const int wv = tid >> 5;
    const int mi = lane & 15;
    const int sel = lane >> 4;

    // --- cooperative load of block-shared constants (zero-padded to kill exec masking) ---
    for (int i = tid; i < 3072; i += 64) {
        int o = i >> 8, l = i & 255;
        sh[OFF_BASE1 + i] = (l < L1) ? base1g[o * L1 + l] : 0.f;
    }
    for (int i = tid; i < FLATN; i += 64) sh[OFF_WEFF + i] = weffg[i];
    for (int i = tid; i < 384; i += 64) {        // [16][24] padded weights
        int row = i / 24, q = i - row * 24;
        sh[OFF_W2F + i] = (row < CCH) ? w2fg[row * 24 + q] : 0.f;
        sh[OFF_W3F + i] = (row < CCH) ? w3fg[row * 24 + q] : 0.f;
    }
    for (int i = tid; i < 24; i += 64) { sh[OFF_WA + i] = wag[i]; sh[OFF_WB + i] = wbg[i]; }
    for (int i = tid; i < 16; i += 64) {
        sh[OFF_B2F + i] = (i < CCH) ? b2fg[i] : 0.f;
        sh[OFF_B3F + i] = (i < CCH) ? b3fg[i] : 0.f;
    }
    if (tid == 0) sh[OFF_WC] = wconstg[0];

    // --- per-wave scratch ---
    float* wb_ = sh + SHC + wv * PW;
    float* abw = wb_;                 // [2][256]
    float* p1w = wb_ + 512;           // [12][P1S]
    float* p2w = p1w + CCH * P1S;     // [12][P2S]
    float* p3w = p2w + CCH * P2S;     // [12][P3S]
    for (int i = lane; i < PW; i += 32) wb_[i] = 0.f;   // zero pads of p1/p2/p3
    __syncthreads();

    // --- hoist per-lane WMMA A fragments (invariant across N-tiles and edges) ---
    v2f aR2[6], aR3[6];
#pragma unroll
    for (int ks = 0; ks < 6; ++ks) {
        int kb = ks * 4 + sel * 2;
        aR2[ks].x = sh[OFF_W2F + mi * 24 + kb];
        aR2[ks].y = sh[OFF_W2F + mi * 24 + kb + 1];
        aR3[ks].x = sh[OFF_W3F + mi * 24 + kb];
        aR3[ks].y = sh[OFF_W3F + mi * 24 + kb + 1];
    }

    for (int rep = 0; rep < EPW; ++rep) {
        const int e = (blockIdx.x * WAVES_PER_BLK + wv) * EPW + rep;
        const int eC = (e < E) ? e : (E - 1);
        const int src = te[2 * eC], dst = te[2 * eC + 1];
        for (int i = lane; i < DN; i += 32) {
            abw[i]      = h[(size_t)src * DN + i];
            abw[DN + i] = h[(size_t)dst * DN + i];
        }
        __syncthreads();

        // --- conv1 (+bn1 folded) + relu + pool: only channels 0/1 vary per edge ---
        for (int j = lane; j < P1L; j += 32) {
            float a0 = abw[2 * j], a1 = abw[2 * j + 1], a2 = abw[2 * j + 2];
            float b0 = abw[DN + 2 * j], b1 = abw[DN + 2 * j + 1], b2 = abw[DN + 2 * j + 2];
#pragma unroll
            for (int o = 0; o < CCH; ++o) {
                float wa0 = sh[OFF_WA + o * 2], wa1 = sh[OFF_WA + o * 2 + 1];
                float wb0 = sh[OFF_WB + o * 2], wb1 = sh[OFF_WB + o * 2 + 1];
                float y0 = sh[OFF_BASE1 + o * 256 + 2 * j]     + wa0 * a0 + wa1 * a1 + wb0 * b0 + wb1 * b1;
                float y1 = sh[OFF_BASE1 + o * 256 + 2 * j + 1] + wa0 * a1 + wa1 * a2 + wb0 * b1 + wb1 * b2;
                p1w[o * P1S + j] = fmaxf(fmaxf(y0, y1), 0.f);   // pool∘relu = relu∘pool
            }
        }
        __syncthreads();

        // --- conv2 as GEMM [16x24]x[24x126] via V_WMMA_F32_16X16X4_F32; shfl pooling ---
#pragma unroll
        for (int nt = 0; nt < 8; ++nt) {
            int n0 = nt * 16;
            v8f acc = {};
#pragma unroll
            for (int ks = 0; ks < 6; ++ks) {
                int kb = ks * 4 + sel * 2;   // even
                v2f b;
                b.x = p1w[(kb >> 1) * P1S + n0 + mi];      // P[q][n] = p1[q>>1][n+(q&1)]
                b.y = p1w[(kb >> 1) * P1S + n0 + mi + 1];
                acc = __builtin_amdgcn_wmma_f32_16x16x4_f32(false, aR2[ks], false, b, (short)0, acc, false, false);
            }
#pragma unroll
            for (int r = 0; r < 8; ++r) {
                int m = r + sel * 8;
                float y = fmaxf(acc[r] + sh[OFF_B2F + m], 0.f);   // padded bias, branch-free
                float yn = __shfl_xor(y, 1, 32);                  // neighbor column (N=lane&15)
                int n = n0 + mi;
                if (((mi & 1) == 0) && (m < CCH) && (n < L2))
                    p2w[m * P2S + (n >> 1)] = fmaxf(y, yn);
            }
        }
        __syncthreads();

        // --- conv3 as GEMM [16x24]x[24x62] via WMMA ---
#pragma unroll
        for (int nt = 0; nt < 4; ++nt) {
            int n0 = nt * 16;
            v8f acc = {};
#pragma unroll
            for (int ks = 0; ks < 6; ++ks) {
                int kb = ks * 4 + sel * 2;
                v2f b;
                b.x = p2w[(kb >> 1) * P2S + n0 + mi];
                b.y = p2w[(kb >> 1) * P2S + n0 + mi + 1];
                acc = __builtin_amdgcn_wmma_f32_16x16x4_f32(false, aR3[ks], false, b, (short)0, acc, false, false);
            }
#pragma unroll
            for (int r = 0; r < 8; ++r) {
                int m = r + sel * 8;
                float y = fmaxf(acc[r] + sh[OFF_B3F + m], 0.f);
                float yn = __shfl_xor(y, 1, 32);
                int n = n0 + mi;
                if (((mi & 1) == 0) && (m < CCH) && (n < L3))
                    p3w[m * P3S + (n >> 1)] = fmaxf(y, yn);
            }
        }
        __syncthreads();

        // --- folded fc1∘bn4∘fc2 dot over flat = concat(p1,p2,p3) per channel, + sigmoid ---
        float partial = 0.f;
        for (int f = lane; f < FLATN; f += 32) {
            int c = f / 221, l = f - c * 221;
            float v;
            if (l < P1L)            v = p1w[c * P1S + l];
            else if (l < P1L + P2L) v = p2w[c * P2S + (l - P1L)];
            else                    v = p3w[c * P3S + (l - P1L - P2L)];
            partial += sh[OFF_WEFF + f] * v;
        }
#pragma unroll
        for (int off = 16; off > 0; off >>= 1) partial += __shfl_down(partial, off, 32);
        if (lane == 0 && e < E)
            out[e] = 1.0f / (1.0f + expf(-(partial + sh[OFF_WC])));
        __syncthreads();
    }
}

// ---------------- host ----------------
extern "C" void kernel_launch(void* const* d_in, const int* in_sizes, int n_in,
                              void* d_out, int out_size, void* d_ws, size_t ws_size,
                              hipStream_t stream) {
    const float* x      = (const float*)d_in[0];
    const int*   ei     = (const int*)  d_in[1];
    const int*   te     = (const int*)  d_in[2];
    const float* Wl     = (const float*)d_in[3];
    const float* bl     = (const float*)d_in[4];
    const float* Wr     = (const float*)d_in[5];
    const float* topkw  = (const float*)d_in[6];
    const float* c1w    = (const float*)d_in[7];
    const float* c1b    = (const float*)d_in[8];
    const float* c2w    = (const float*)d_in[9];
    const float* c2b    = (const float*)d_in[10];
    const float* c3w    = (const float*)d_in[11];
    const float* c3b    = (const float*)d_in[12];
    const float* g1 = (const float*)d_in[13]; const float* bb1 = (const float*)d_in[14];
    const float* g2 = (const float*)d_in[15]; const float* bb2 = (const float*)d_in[16];
    const float* g3 = (const float*)d_in[17]; const float* bb3 = (const float*)d_in[18];
    const float* g4 = (const float*)d_in[19]; const float* bb4 = (const float*)d_in[20];
    const float* fc1w = (const float*)d_in[21]; const float* fc1b = (const float*)d_in[22];
    const float* fc2w = (const float*)d_in[23]; const float* fc2b = (const float*)d_in[24];
    float* out = (float*)d_out;

    const int Eg     = in_sizes[1] / 2;
    const int E      = in_sizes[2] / 2;
    const int nNodes = in_sizes[0] / DN;

    // workspace layout (floats) — ~8.8 MB total
    float* ws = (float*)d_ws;
    size_t oAGG   = 0;
    size_t oDEG   = oAGG + (size_t)nNodes * DN;
    size_t oINV   = oDEG + nNodes;
    size_t oH     = oINV + nNodes;
    size_t oTOPK  = oH + (size_t)nNodes * DN;
    size_t oBASE1 = oTOPK + 10 * DN;
    size_t oWA    = oBASE1 + 12 * L1;
    size_t oWB    = oWA + 24;
    size_t oW2F   = oWB + 24;
    size_t oB2F   = oW2F + 288;
    size_t oW3F   = oB2F + 12;
    size_t oB3F   = oW3F + 288;
    size_t oWEFF  = oB3F + 12;
    size_t oWC    = oWEFF + FLATN;
    size_t oAF    = oWC + 1;
    size_t oWF    = oAF + (size_t)nNodes * 512;

    // 1) zero agg + deg
    long nz = (long)nNodes * DN + nNodes;
    zero_kernel<<<512, 256, 0, stream>>>(ws + oAGG, nz);
    // 2) scatter-add aggregation (GLOBAL_ATOMIC_ADD_F32)
    scatter_kernel<<<Eg, DN, 0, stream>>>(ei, x, ws + oAGG, ws + oDEG, Eg);
    // 3) 1/max(deg,1)
    invdeg_kernel<<<(nNodes + 255) / 256, 256, 0, stream>>>(ws + oDEG, ws + oINV, nNodes);
    // 4) pack branch-free GEMM operands
    long nAF = (long)nNodes * 512;
    pack_af<<<(int)((nAF + 255) / 256), 256, 0, stream>>>(ws + oAGG, ws + oINV, x, ws + oAF, nNodes);
    pack_wf<<<512, 256, 0, stream>>>(Wl, Wr, ws + oWF);
    // 5) h = relu(af @ wf^T + bl) — fp32 WMMA, one wave per 16x16 tile
    dim3 g4d((nNodes + 15) / 16, DN / 16);
    hgemm_wmma<<<g4d, 32, 0, stream>>>(ws + oAF, ws + oWF, bl, ws + oH, nNodes);
    // 6) top-10 pooling
    topk_kernel<<<1, 256, 0, stream>>>(ws + oH, topkw, ws + oTOPK, nNodes);
    // 7) fold constants: conv1 shared base, BN folds, fc1∘bn4∘fc2 collapse
    prep_kernel<<<1, 256, 0, stream>>>(ws + oTOPK, c1w, c1b, c2w, c2b, c3w, c3b,
                                       g1, bb1, g2, bb2, g3, bb3, g4, bb4,
                                       fc1w, fc1b, fc2w, fc2b,
                                       ws + oBASE1, ws + oWA, ws + oWB, ws + oW2F, ws + oB2F,
                                       ws + oW3F, ws + oB3F, ws + oWEFF, ws + oWC);
    // 8) fused per-edge CNN + folded fc + sigmoid (one wave32 per edge-group, WMMA conv2/conv3)
    int edgesPerBlk = WAVES_PER_BLK * EPW;
    int blocks = (E + edgesPerBlk - 1) / edgesPerBlk;
    size_t shbytes = (size_t)LDS_FLOATS * sizeof(float);   // ~52 KB per 64-thread block
    edge_cnn_kernel<<<blocks, 32 * WAVES_PER_BLK, shbytes, stream>>>(
        te, ws + oH, ws + oBASE1, ws + oWA, ws + oWB, ws + oW2F, ws + oB2F,
        ws + oW3F, ws + oB3F, ws + oWEFF, ws + oWC, out, E);
}